// Encoder_1511828488869
// MI455X (gfx1250) — compile-verified
//
#include <hip/hip_runtime.h>

typedef __attribute__((ext_vector_type(2))) float v2f;
typedef __attribute__((ext_vector_type(8))) float v8f;

// Problem constants (from reference): L=4, M=64, F=128, E=128, RAW=641, H=320
#define KP   644      // RAW padded to multiple of 4 (rows 641..643 zero)
#define HH   320      // MLP hidden
#define MM   64       // memory dim
#define EE   128      // embedding dim
#define MF   256      // mem_flat width (L*M)

// Weight layouts: paired-K interleave  Wp[k/2][col][2]  so a WMMA B-fragment
// (rows k, k+1 at one column) is a single aligned 8-byte load.

// ---------------- init / small kernels ----------------

__global__ void k_zero(float* acc, long accN, unsigned* tb, int* cnt, int n) {
  long i = (long)blockIdx.x * blockDim.x + threadIdx.x;
  long stride = (long)gridDim.x * blockDim.x;
  for (long j = i; j < accN; j += stride) acc[j] = 0.0f;
  for (long j = i; j < n; j += stride) { tb[j] = 0u; cnt[j] = 0; }
}

__global__ void k_transpose(const float* __restrict__ W1, const float* __restrict__ W2,
                            const float* __restrict__ We, const float* __restrict__ Ws,
                            float* __restrict__ W1P, float* __restrict__ W2P,
                            float* __restrict__ WeP, float* __restrict__ WsP) {
  const long n1 = (long)KP * HH;          // W1P [322][320][2]
  const long n2 = (long)HH * MM;          // W2P [160][64][2]
  const long n3 = (long)MF * MM;          // WeP [128][64][2]
  const long n4 = (long)MM * EE;          // WsP [32][128][2]
  long total = n1 + n2 + n3 + n4;
  long i = (long)blockIdx.x * blockDim.x + threadIdx.x;
  long stride = (long)gridDim.x * blockDim.x;
  for (long j = i; j < total; j += stride) {
    if (j < n1) {
      long k2 = j / (HH * 2), rem = j - k2 * (HH * 2);
      long h = rem >> 1, k = 2 * k2 + (rem & 1);
      W1P[j] = (k < 641) ? W1[h * 641 + k] : 0.0f;
    } else if (j < n1 + n2) {
      long q = j - n1;
      long k2 = q / (MM * 2), rem = q - k2 * (MM * 2);
      long m = rem >> 1, k = 2 * k2 + (rem & 1);
      W2P[q] = W2[m * HH + k];
    } else if (j < n1 + n2 + n3) {
      long q = j - n1 - n2;
      long k2 = q / (MM * 2), rem = q - k2 * (MM * 2);
      long m = rem >> 1, k = 2 * k2 + (rem & 1);
      WeP[q] = We[m * MF + k];
    } else {
      long q = j - n1 - n2 - n3;
      long k2 = q / (EE * 2), rem = q - k2 * (EE * 2);
      long e = rem >> 1, k = 2 * k2 + (rem & 1);
      WsP[q] = Ws[e * MM + k];
    }
  }
}

__global__ void k_seg(const float* __restrict__ ts, const int* __restrict__ src,
                      unsigned* tb, int* cnt, int nB) {
  int i = blockIdx.x * blockDim.x + threadIdx.x;
  if (i < nB) {
    int s = src[i];
    atomicMax(&tb[s], __float_as_uint(ts[i]));  // ts >= 0 -> bit order == float order
    atomicAdd(&cnt[s], 1);
  }
}

__global__ void k_tsafe(const unsigned* __restrict__ tb, const int* __restrict__ cnt,
                        const float* __restrict__ last, float* tsafe, int n) {
  int i = blockIdx.x * blockDim.x + threadIdx.x;
  if (i < n) tsafe[i] = (cnt[i] > 0) ? __uint_as_float(tb[i]) : last[i];
}

// ---------------- per-event WMMA GEMM + scatter ----------------
// y[b] = W1 @ raw[b]  (16 events/block), then Acc[src,l,:] += w[b,l]*y[b]
// 5 waves x 4 col-tiles: each A fragment (1 LDS b64) feeds 4 WMMAs.
__global__ __launch_bounds__(160) void k_event(
    const float* __restrict__ mem, const float* __restrict__ feat,
    const float* __restrict__ ts, const int* __restrict__ src, const int* __restrict__ tgt,
    const float* __restrict__ tsafe, const float* __restrict__ lambs,
    const float* __restrict__ W1Pf, float* __restrict__ Acc, int nB) {
  __shared__ float raw[16 * KP];   // 644 % 64 == 4 -> conflict-free A reads
  __shared__ float wl[16][4];
  __shared__ int ssrc[16];
  __shared__ int stgt[16];
  int t = threadIdx.x;
  int base = blockIdx.x * 16;
  if (t < 16) {
    int e = base + t;
    int ok = (e < nB);
    int sn = ok ? src[e] : 0;
    int tn = ok ? tgt[e] : 0;
    ssrc[t] = sn; stgt[t] = tn;
    float dt = ok ? (tsafe[sn] - ts[e]) : 0.0f;
#pragma unroll
    for (int l = 0; l < 4; l++) wl[t][l] = ok ? __expf(-dt * lambs[l]) : 0.0f;
  }
  __syncthreads();
  for (int i = t; i < 16 * KP; i += 160) {
    int e = i / KP, k = i - e * KP;
    float v;
    if (k < 256)       v = mem[(size_t)ssrc[e] * MF + k];
    else if (k < 512)  v = mem[(size_t)stgt[e] * MF + (k - 256)];
    else if (k < 640)  v = feat[(size_t)stgt[e] * 128 + (k - 512)];
    else               v = (k == 640) ? 1.0f : 0.0f;
    raw[i] = v;
  }
  __syncthreads();

  const v2f* __restrict__ W1P = (const v2f*)W1Pf;  // [KP/2][HH] v2f
  int wid = t >> 5, lane = t & 31;
  int mrow = lane & 15, koff = (lane >> 4) * 2;  // A: lanes0-15 K={0,1}, lanes16-31 K={2,3}
  int c0 = wid * 64;                              // 5 waves x 64 cols = 320
  v8f a0 = {}, a1 = {}, a2 = {}, a3 = {};
  for (int kk = 0; kk < KP; kk += 4) {
    v2f A; A.x = raw[mrow * KP + kk + koff]; A.y = raw[mrow * KP + kk + koff + 1];
    size_t bofs = (size_t)((kk + koff) >> 1) * HH + c0 + mrow;
    v2f B0 = W1P[bofs];
    v2f B1 = W1P[bofs + 16];
    v2f B2 = W1P[bofs + 32];
    v2f B3 = W1P[bofs + 48];
    a0 = __builtin_amdgcn_wmma_f32_16x16x4_f32(false, A, false, B0, (short)0, a0, false, false);
    a1 = __builtin_amdgcn_wmma_f32_16x16x4_f32(false, A, false, B1, (short)0, a1, false, false);
    a2 = __builtin_amdgcn_wmma_f32_16x16x4_f32(false, A, false, B2, (short)0, a2, false, false);
    a3 = __builtin_amdgcn_wmma_f32_16x16x4_f32(false, A, false, B3, (short)0, a3, false, false);
  }
  // C/D layout: VGPR r -> row r (lanes 0-15) / row r+8 (lanes 16-31), col = lane&15
  int rbase = (lane >> 4) * 8;
#pragma unroll
  for (int r = 0; r < 8; r++) {
    int erow = rbase + r;
    int node = ssrc[erow];
    float y0 = a0[r], y1 = a1[r], y2 = a2[r], y3 = a3[r];
#pragma unroll
    for (int l = 0; l < 4; l++) {
      float w = wl[erow][l];
      float* dst = Acc + ((size_t)node * 4 + l) * HH + c0 + mrow;
      atomicAdd(dst,      w * y0);
      atomicAdd(dst + 16, w * y1);
      atomicAdd(dst + 32, w * y2);
      atomicAdd(dst + 48, w * y3);
    }
  }
}

// ---------------- msg = W2 @ relu(Acc + b1) + b2 ; memory decay update ----------------
__global__ __launch_bounds__(128) void k_msg(
    const float* __restrict__ Acc, const float* __restrict__ b1,
    const float* __restrict__ W2Pf, const float* __restrict__ b2,
    const float* __restrict__ mem, const float* __restrict__ tsafe,
    const float* __restrict__ last, const int* __restrict__ cnt,
    const float* __restrict__ lambs, float* __restrict__ dyn, int NL) {
  const int HS = 324;  // 324 % 64 == 4
  __shared__ float Hs[16 * HS];
  __shared__ float dec[16];
  __shared__ int hasl[16];
  int t = threadIdx.x;
  int rowbase = blockIdx.x * 16;
  if (t < 16) {
    int row = rowbase + t; if (row >= NL) row = NL - 1;
    int node = row >> 2, l = row & 3;
    hasl[t] = (cnt[node] > 0);
    dec[t] = __expf(-(tsafe[node] - last[node]) * lambs[l]);
  }
  for (int i = t; i < 16 * HH; i += 128) {
    int r = i / HH, k = i - r * HH;
    int row = rowbase + r; if (row >= NL) row = NL - 1;
    float v = Acc[(size_t)row * HH + k] + b1[k];
    Hs[r * HS + k] = v > 0.0f ? v : 0.0f;
  }
  __syncthreads();
  const v2f* __restrict__ W2P = (const v2f*)W2Pf;  // [HH/2][MM] v2f
  int wid = t >> 5, lane = t & 31;
  int mrow = lane & 15, koff = (lane >> 4) * 2, c0 = wid * 16;
  v8f acc = {};
  for (int kk = 0; kk < HH; kk += 4) {
    v2f A; A.x = Hs[mrow * HS + kk + koff]; A.y = Hs[mrow * HS + kk + koff + 1];
    v2f Bv = W2P[(size_t)((kk + koff) >> 1) * MM + c0 + mrow];
    acc = __builtin_amdgcn_wmma_f32_16x16x4_f32(false, A, false, Bv, (short)0, acc, false, false);
  }
  int rb = (lane >> 4) * 8, c = c0 + mrow;
  float bias = b2[c];
#pragma unroll
  for (int r = 0; r < 8; r++) {
    int m = rb + r, row = rowbase + m;
    if (row < NL) {
      int node = row >> 2, l = row & 3;
      float mv = mem[(size_t)row * MM + c];
      float msg = acc[r] + bias;
      dyn[(size_t)node * MF + l * MM + c] = mv * dec[m] + (hasl[m] ? msg : 0.0f);
    }
  }
}

// ---------------- embedding fusion: leaky(We@dyn+be) -> leaky(Ws@e+bs) -> blend ----------------
__global__ __launch_bounds__(256) void k_emb(
    const float* __restrict__ dyn, const float* __restrict__ WePf, const float* __restrict__ be,
    const float* __restrict__ WsPf, const float* __restrict__ bs,
    const float* __restrict__ stat, const float* __restrict__ theta,
    float* __restrict__ out, int N) {
  const int DS = 260, ES = 68;  // both % 64 == 4
  __shared__ float D[16 * DS];
  __shared__ float Eld[16 * ES];
  int t = threadIdx.x;
  int nb = blockIdx.x * 16;
  for (int i = t; i < 16 * MF; i += 256) {
    int r = i >> 8, k = i & 255;
    int node = nb + r; if (node >= N) node = N - 1;
    D[r * DS + k] = dyn[(size_t)node * MF + k];
  }
  __syncthreads();
  const v2f* __restrict__ WeP = (const v2f*)WePf;  // [MF/2][MM] v2f
  const v2f* __restrict__ WsP = (const v2f*)WsPf;  // [MM/2][EE] v2f
  int wid = t >> 5, lane = t & 31, mrow = lane & 15, koff = (lane >> 4) * 2;
  if (wid < 4) {  // wave-uniform branch: EXEC all-ones inside
    int c0 = wid * 16;
    v8f acc = {};
    for (int kk = 0; kk < MF; kk += 4) {
      v2f A; A.x = D[mrow * DS + kk + koff]; A.y = D[mrow * DS + kk + koff + 1];
      v2f Bv = WeP[(size_t)((kk + koff) >> 1) * MM + c0 + mrow];
      acc = __builtin_amdgcn_wmma_f32_16x16x4_f32(false, A, false, Bv, (short)0, acc, false, false);
    }
    int rb = (lane >> 4) * 8, c = c0 + mrow;
    float bias = be[c];
#pragma unroll
    for (int r = 0; r < 8; r++) {
      float v = acc[r] + bias;
      Eld[(rb + r) * ES + c] = v > 0.0f ? v : 0.01f * v;
    }
  }
  __syncthreads();
  {
    int c0 = wid * 16;  // 8 waves cover 128 output cols
    v8f acc = {};
    for (int kk = 0; kk < MM; kk += 4) {
      v2f A; A.x = Eld[mrow * ES + kk + koff]; A.y = Eld[mrow * ES + kk + koff + 1];
      v2f Bv = WsP[(size_t)((kk + koff) >> 1) * EE + c0 + mrow];
      acc = __builtin_amdgcn_wmma_f32_16x16x4_f32(false, A, false, Bv, (short)0, acc, false, false);
    }
    float th = theta[0];
    int rb = (lane >> 4) * 8, c = c0 + mrow;
    float bias = bs[c];
#pragma unroll
    for (int r = 0; r < 8; r++) {
      int node = nb + rb + r;
      if (node < N) {
        float v = acc[r] + bias;
        v = v > 0.0f ? v : 0.01f * v;
        out[(size_t)node * EE + c] = th * stat[(size_t)node * EE + c] + (1.0f - th) * v;
      }
    }
  }
}

// ---------------- launcher ----------------

extern "C" void kernel_launch(void* const* d_in, const int* in_sizes, int n_in,
                              void* d_out, int out_size, void* d_ws, size_t ws_size,
                              hipStream_t stream) {
  (void)n_in; (void)out_size; (void)ws_size;
  const float* node_features = (const float*)d_in[0];
  const float* memory        = (const float*)d_in[1];
  const float* last_update   = (const float*)d_in[2];
  const float* lambs         = (const float*)d_in[3];
  const float* theta         = (const float*)d_in[4];
  const float* static_emb    = (const float*)d_in[5];
  const float* W1 = (const float*)d_in[6];
  const float* b1 = (const float*)d_in[7];
  const float* W2 = (const float*)d_in[8];
  const float* b2 = (const float*)d_in[9];
  const float* We = (const float*)d_in[10];
  const float* be = (const float*)d_in[11];
  const float* Ws = (const float*)d_in[12];
  const float* bs = (const float*)d_in[13];
  const float* timestamps = (const float*)d_in[14];
  const int* src = (const int*)d_in[15];
  const int* tgt = (const int*)d_in[16];
  float* out = (float*)d_out;

  int Nn = in_sizes[2];   // N (last_update is [N])
  int nB = in_sizes[14];  // B (timestamps is [B])

  char* w = (char*)d_ws;
  size_t off = 0;
  auto take = [&](size_t bytes) -> char* {
    char* p = w + off;
    off = (off + bytes + 255) & ~(size_t)255;
    return p;
  };
  float*    Acc   = (float*)take((size_t)Nn * 4 * HH * sizeof(float));  // ~102 MB (L2-resident)
  float*    dyn   = (float*)take((size_t)Nn * MF * sizeof(float));      // ~20.5 MB
  float*    W1P   = (float*)take((size_t)KP * HH * sizeof(float));
  float*    W2P   = (float*)take((size_t)HH * MM * sizeof(float));
  float*    WeP   = (float*)take((size_t)MF * MM * sizeof(float));
  float*    WsP   = (float*)take((size_t)MM * EE * sizeof(float));
  unsigned* tb    = (unsigned*)take((size_t)Nn * sizeof(unsigned));
  int*      cnt   = (int*)take((size_t)Nn * sizeof(int));
  float*    tsafe = (float*)take((size_t)Nn * sizeof(float));

  k_zero<<<2048, 256, 0, stream>>>(Acc, (long)Nn * 4 * HH, tb, cnt, Nn);
  long ttotal = (long)KP * HH + (long)HH * MM + (long)MF * MM + (long)MM * EE;
  k_transpose<<<(int)((ttotal + 255) / 256), 256, 0, stream>>>(W1, W2, We, Ws, W1P, W2P, WeP, WsP);
  k_seg<<<(nB + 255) / 256, 256, 0, stream>>>(timestamps, src, tb, cnt, nB);
  k_tsafe<<<(Nn + 255) / 256, 256, 0, stream>>>(tb, cnt, last_update, tsafe, Nn);
  k_event<<<(nB + 15) / 16, 160, 0, stream>>>(memory, node_features, timestamps, src, tgt,
                                              tsafe, lambs, W1P, Acc, nB);
  k_msg<<<(Nn * 4 + 15) / 16, 128, 0, stream>>>(Acc, b1, W2P, b2, memory, tsafe, last_update,
                                                cnt, lambs, dyn, Nn * 4);
  k_emb<<<(Nn + 15) / 16, 256, 0, stream>>>(dyn, WeP, be, WsP, bs, static_emb, theta, out, Nn);
}